// PointNetSetAbstraction_41540923687664
// MI455X (gfx1250) — compile-verified
//
#include <hip/hip_runtime.h>
#include <hip/hip_bf16.h>
#include <float.h>

// ---------------- problem constants ----------------
#define BATCH   16
#define NPTS    4096
#define NPOINT  1024
#define NSAMPLE 32
#define CIN     64          // input feature channels
#define K0PAD   68          // 3 + 64 = 67, padded to multiple of 4
#define C1      64
#define C2      64
#define C3      128
#define MROWS   (BATCH * NPOINT * NSAMPLE)   // 524288 gemm rows
#define BN_EPS  1e-5f

typedef float v2f __attribute__((ext_vector_type(2)));
typedef float v8f __attribute__((ext_vector_type(8)));

// ---------------- workspace layout (bytes) ----------------
#define WS_IDX   ((size_t)0)                           // int32 idx, 2MB
#define WS_STATS ((size_t)4 << 20)                     // 3 * [sum128 sq128 s128 t128]
#define WS_W0P   (((size_t)4 << 20) + (64u << 10))     // padded w0 (64 x 68)
#define WS_R0    ((size_t)8 << 20)                     // x0 (M x 68), later y2 (M x 64)
#define WS_R1    ((size_t)168 << 20)                   // y1 (M x 64), later y3 (M x 128)

// =====================================================================
// 0) zero the BN stats accumulators (atomics target)
// =====================================================================
__global__ void zero_stats_kernel(float* stats) {
    int i = blockIdx.x * blockDim.x + threadIdx.x;
    if (i < 3 * 512) stats[i] = 0.0f;
}

// =====================================================================
// 1) Farthest point sampling. One block per batch, 1024 threads.
//    xyz cached in LDS (48KB); running min-distances in registers
//    (4 points per thread). argmax w/ first-index tie-break.
// =====================================================================
__global__ void __launch_bounds__(1024) fps_kernel(
    const float* __restrict__ xyz, float* __restrict__ new_xyz)
{
    __shared__ float sx[NPTS], sy[NPTS], sz[NPTS];     // 48KB
    __shared__ float rv[32];
    __shared__ int   ri[32];
    __shared__ int   sfar;

    const int b = blockIdx.x;
    const int t = threadIdx.x;
    const float* base = xyz + (size_t)b * NPTS * 3;

    for (int i = t; i < NPTS; i += 1024) {
        sx[i] = base[3 * i + 0];
        sy[i] = base[3 * i + 1];
        sz[i] = base[3 * i + 2];
    }
    __syncthreads();

    // thread t owns points t, t+1024, t+2048, t+3072 (increasing order)
    float d0 = 1e10f, d1 = 1e10f, d2r = 1e10f, d3 = 1e10f;
    int farthest = 0;

    for (int s = 0; s < NPOINT; ++s) {
        const float cx = sx[farthest], cy = sy[farthest], cz = sz[farthest];
        if (t == 0) {
            float* o = new_xyz + ((size_t)b * NPOINT + s) * 3;
            o[0] = cx; o[1] = cy; o[2] = cz;
        }
        float bestv = -1.0f; int besti = 0;
        {
            int i = t;
            float dx = sx[i]-cx, dy = sy[i]-cy, dz = sz[i]-cz;
            d0 = fminf(d0, dx*dx + dy*dy + dz*dz);
            if (d0 > bestv) { bestv = d0; besti = i; }
            i = t + 1024;
            dx = sx[i]-cx; dy = sy[i]-cy; dz = sz[i]-cz;
            d1 = fminf(d1, dx*dx + dy*dy + dz*dz);
            if (d1 > bestv) { bestv = d1; besti = i; }
            i = t + 2048;
            dx = sx[i]-cx; dy = sy[i]-cy; dz = sz[i]-cz;
            d2r = fminf(d2r, dx*dx + dy*dy + dz*dz);
            if (d2r > bestv) { bestv = d2r; besti = i; }
            i = t + 3072;
            dx = sx[i]-cx; dy = sy[i]-cy; dz = sz[i]-cz;
            d3 = fminf(d3, dx*dx + dy*dy + dz*dz);
            if (d3 > bestv) { bestv = d3; besti = i; }
        }
        // wave32 reduction (lower index wins ties)
        #pragma unroll
        for (int off = 16; off > 0; off >>= 1) {
            float ov = __shfl_down(bestv, off, 32);
            int   oi = __shfl_down(besti, off, 32);
            if (ov > bestv || (ov == bestv && oi < besti)) { bestv = ov; besti = oi; }
        }
        const int wid = t >> 5;
        if ((t & 31) == 0) { rv[wid] = bestv; ri[wid] = besti; }
        __syncthreads();
        if (t == 0) {
            float v = rv[0]; int ix = ri[0];
            for (int w = 1; w < 32; ++w)
                if (rv[w] > v || (rv[w] == v && ri[w] < ix)) { v = rv[w]; ix = ri[w]; }
            sfar = ix;
        }
        __syncthreads();
        farthest = sfar;
        __syncthreads();
    }
}

// =====================================================================
// 2) kNN: one wave per (b, centroid). d2 = |c|^2 + |p|^2 - 2 c.p in LDS,
//    then 32 rounds of wave-wide argmin with first-index tie-break.
// =====================================================================
__global__ void __launch_bounds__(32) knn_kernel(
    const float* __restrict__ xyz, const float* __restrict__ new_xyz,
    int* __restrict__ idx_out)
{
    __shared__ float d2s[NPTS];
    const int gid  = blockIdx.x;               // b*NPOINT + m
    const int b    = gid >> 10;
    const int lane = threadIdx.x;
    const float* pb = xyz + (size_t)b * NPTS * 3;

    const float cx = new_xyz[(size_t)gid * 3 + 0];
    const float cy = new_xyz[(size_t)gid * 3 + 1];
    const float cz = new_xyz[(size_t)gid * 3 + 2];
    const float sqn = cx * cx + cy * cy + cz * cz;

    for (int i = lane; i < NPTS; i += 32) {
        float px = pb[3 * i + 0], py = pb[3 * i + 1], pz = pb[3 * i + 2];
        float sq = px * px + py * py + pz * pz;
        d2s[i] = sqn + sq - 2.0f * (cx * px + cy * py + cz * pz);
    }
    __syncthreads();

    int* ob = idx_out + (size_t)gid * NSAMPLE;
    for (int k = 0; k < NSAMPLE; ++k) {
        float bv = FLT_MAX; int bi = 0x7fffffff;
        for (int i = lane; i < NPTS; i += 32) {
            float v = d2s[i];
            if (v < bv || (v == bv && i < bi)) { bv = v; bi = i; }
        }
        #pragma unroll
        for (int off = 16; off > 0; off >>= 1) {
            float ov = __shfl_down(bv, off, 32);
            int   oi = __shfl_down(bi, off, 32);
            if (ov < bv || (ov == bv && oi < bi)) { bv = ov; bi = oi; }
        }
        bi = __shfl(bi, 0, 32);
        if (lane == 0) { ob[k] = bi; d2s[bi] = FLT_MAX; }
        __syncthreads();
    }
}

// =====================================================================
// 3) pad w0 (64x67) -> (64x68) with trailing zero column
// =====================================================================
__global__ void pad_w0_kernel(const float* __restrict__ w0, float* __restrict__ w0p) {
    int i = blockIdx.x * blockDim.x + threadIdx.x;
    if (i >= C1 * K0PAD) return;
    int r = i / K0PAD, c = i % K0PAD;
    w0p[i] = (c < 67) ? w0[r * 67 + c] : 0.0f;
}

// =====================================================================
// 4) materialize x0 (M x 68): one wave per row, lane-strided copy
// =====================================================================
__global__ void __launch_bounds__(256) build_x0_kernel(
    const float* __restrict__ xyz, const float* __restrict__ points,
    const float* __restrict__ new_xyz, const int* __restrict__ idx,
    float* __restrict__ x0)
{
    const int row  = blockIdx.x * 8 + (threadIdx.x >> 5);
    const int lane = threadIdx.x & 31;
    if (row >= MROWS) return;
    const size_t bm = (size_t)row >> 5;        // b*NPOINT + m
    const int b = (int)(bm >> 10);
    const int j = idx[row];
    const float* p = xyz + ((size_t)b * NPTS + j) * 3;
    const float* c = new_xyz + bm * 3;
    const float* f = points + ((size_t)b * NPTS + j) * CIN;
    float* o = x0 + (size_t)row * K0PAD;
    if (lane < 3) o[lane] = p[lane] - c[lane];
    if (lane == 3) o[67] = 0.0f;
    o[3 + lane]      = f[lane];
    o[3 + lane + 32] = f[lane + 32];
}

// =====================================================================
// 5) WMMA fp32 GEMM + bias + fused per-channel BN-stats accumulation,
//    with optional fused input-side BN+ReLU (previous layer's s/t).
//    Y[M x N] = relu_bn(X)[M x K] * W[N x K]^T + bias.
//    - B fragments and (s,t) fragments are wave-invariant: registers.
//    - 2 M-tiles per iteration, two interleaved WMMA chains.
//    A (16x4): lane<16 holds row lane, K {kb,kb+1}; lane>=16 K {kb+2,kb+3}.
//    C/D:      VGPR v, lane<16 -> (m0+v, n0+lane); lane>=16 -> (m0+v+8, ...).
// =====================================================================
template <int KB, bool FUSE>   // KB = K/4 (16 or 17)
__global__ void __launch_bounds__(256) gemm_wmma_kernel(
    const float* __restrict__ X, int ldx,
    const float* __restrict__ W, int ldw,
    const float* __restrict__ bias,
    const float* __restrict__ sprev,   // scale per input channel (FUSE)
    const float* __restrict__ tprev,   // shift per input channel (FUSE)
    float* __restrict__ Y, int Ncols,
    float* __restrict__ ssum, float* __restrict__ ssq)
{
    const int lane = threadIdx.x & 31;
    const int wave = threadIdx.x >> 5;         // 0..7
    const int kh   = lane >> 4;                // K-half select (0 or 1)
    const int l16  = lane & 15;
    const int n0   = blockIdx.y * 16;
    const int ch   = n0 + l16;                 // this lane's output channel
    const int nwaves = gridDim.x * 8;
    const int wgid   = blockIdx.x * 8 + wave;
    const int mtiles = MROWS >> 4;

    // ---- preload wave-invariant B fragments (W row 'ch', all K blocks) ----
    v2f bfrag[KB];
    {
        const float* wrow = W + (size_t)ch * ldw + 2 * kh;
        #pragma unroll
        for (int i = 0; i < KB; ++i) {
            bfrag[i].x = wrow[4 * i];
            bfrag[i].y = wrow[4 * i + 1];
        }
    }
    // ---- preload per-input-channel BN scale/shift fragments ----
    v2f sf[KB], tf[KB];
    if (FUSE) {
        #pragma unroll
        for (int i = 0; i < KB; ++i) {
            const int k = 4 * i + 2 * kh;
            sf[i].x = sprev[k];     sf[i].y = sprev[k + 1];
            tf[i].x = tprev[k];     tf[i].y = tprev[k + 1];
        }
    }

    float acc_s = 0.0f, acc_q = 0.0f;
    const float bv = bias[ch];

    for (int mt = wgid * 2; mt < mtiles; mt += nwaves * 2) {
        const int m0 = mt << 4;
        const float* xr0 = X + (size_t)(m0 + l16) * ldx + 2 * kh;
        const float* xr1 = xr0 + (size_t)16 * ldx;

        v2f a0[KB], a1[KB];
        #pragma unroll
        for (int i = 0; i < KB; ++i) {
            a0[i].x = xr0[4 * i]; a0[i].y = xr0[4 * i + 1];
        }
        #pragma unroll
        for (int i = 0; i < KB; ++i) {
            a1[i].x = xr1[4 * i]; a1[i].y = xr1[4 * i + 1];
        }
        if (FUSE) {
            #pragma unroll
            for (int i = 0; i < KB; ++i) {
                a0[i].x = fmaxf(fmaf(a0[i].x, sf[i].x, tf[i].x), 0.0f);
                a0[i].y = fmaxf(fmaf(a0[i].y, sf[i].y, tf[i].y), 0.0f);
                a1[i].x = fmaxf(fmaf(a1[i].x, sf[i].x, tf[i].x), 0.0f);
                a1[i].y = fmaxf(fmaf(a1[i].y, sf[i].y, tf[i].y), 0.0f);
            }
        }

        v8f c0 = {0.f, 0.f, 0.f, 0.f, 0.f, 0.f, 0.f, 0.f};
        v8f c1 = {0.f, 0.f, 0.f, 0.f, 0.f, 0.f, 0.f, 0.f};
        #pragma unroll
        for (int i = 0; i < KB; ++i) {
            c0 = __builtin_amdgcn_wmma_f32_16x16x4_f32(
                     false, a0[i], false, bfrag[i], (short)0, c0, false, false);
            c1 = __builtin_amdgcn_wmma_f32_16x16x4_f32(
                     false, a1[i], false, bfrag[i], (short)0, c1, false, false);
        }

        float* yb0 = Y + (size_t)(m0 + kh * 8) * Ncols + ch;
        float* yb1 = yb0 + (size_t)16 * Ncols;
        #pragma unroll
        for (int v = 0; v < 8; ++v) {
            float val0 = c0[v] + bv;
            float val1 = c1[v] + bv;
            acc_s += val0 + val1;
            acc_q += val0 * val0 + val1 * val1;
            yb0[(size_t)v * Ncols] = val0;
            yb1[(size_t)v * Ncols] = val1;
        }
    }
    atomicAdd(&ssum[ch], acc_s);
    atomicAdd(&ssq[ch], acc_q);
}

// =====================================================================
// 6) finalize BN stats -> per-channel scale/shift
// =====================================================================
__global__ void finalize_stats_kernel(
    const float* __restrict__ ssum, const float* __restrict__ ssq,
    const float* __restrict__ g, const float* __restrict__ beta,
    float* __restrict__ s, float* __restrict__ t, int Ncols, float invM)
{
    int c = threadIdx.x;
    if (c >= Ncols) return;
    float mean = ssum[c] * invM;
    float var  = ssq[c] * invM - mean * mean;
    float sc   = g[c] * rsqrtf(var + BN_EPS);
    s[c] = sc;
    t[c] = beta[c] - mean * sc;
}

// =====================================================================
// 7) layer-3 BN+ReLU fused with max over NSAMPLE + transpose to (B,128,NP)
// =====================================================================
__global__ void __launch_bounds__(128) maxpool_out_kernel(
    const float* __restrict__ Y3, const float* __restrict__ s,
    const float* __restrict__ t, float* __restrict__ out)
{
    const int bm = blockIdx.x;                 // b*NPOINT + m
    const int c  = threadIdx.x;                // 0..127
    const float sc = s[c], tc = t[c];
    const float* base = Y3 + (size_t)bm * NSAMPLE * C3 + c;
    float mv = -FLT_MAX;
    #pragma unroll 4
    for (int si = 0; si < NSAMPLE; ++si) {
        float v = fmaxf(fmaf(base[(size_t)si * C3], sc, tc), 0.0f);
        mv = fmaxf(mv, v);
    }
    const int b = bm >> 10, m = bm & 1023;
    out[((size_t)b * C3 + c) * NPOINT + m] = mv;
}

// =====================================================================
// launch
// =====================================================================
extern "C" void kernel_launch(void* const* d_in, const int* in_sizes, int n_in,
                              void* d_out, int out_size, void* d_ws, size_t ws_size,
                              hipStream_t stream) {
    const float* xyz    = (const float*)d_in[0];
    const float* points = (const float*)d_in[1];
    const float* w0 = (const float*)d_in[2];
    const float* b0 = (const float*)d_in[3];
    const float* g0 = (const float*)d_in[4];
    const float* be0= (const float*)d_in[5];
    const float* w1 = (const float*)d_in[6];
    const float* b1 = (const float*)d_in[7];
    const float* g1 = (const float*)d_in[8];
    const float* be1= (const float*)d_in[9];
    const float* w2 = (const float*)d_in[10];
    const float* b2 = (const float*)d_in[11];
    const float* g2 = (const float*)d_in[12];
    const float* be2= (const float*)d_in[13];

    float* out_xyz = (float*)d_out;                                   // (B,NPOINT,3)
    float* out_pts = (float*)d_out + (size_t)BATCH * NPOINT * 3;      // (B,128,NPOINT)

    char* ws = (char*)d_ws;
    int*   idx   = (int*)  (ws + WS_IDX);
    float* stats = (float*)(ws + WS_STATS);        // 3 * [sum128 sq128 s128 t128]
    float* w0p   = (float*)(ws + WS_W0P);
    float* x0    = (float*)(ws + WS_R0);           // M x 68
    float* y1    = (float*)(ws + WS_R1);           // M x 64
    float* y2    = (float*)(ws + WS_R0);           // M x 64 (reuses x0)
    float* y3    = (float*)(ws + WS_R1);           // M x 128 (overwrites dead y1)

    float* st0 = stats + 0 * 512;
    float* st1 = stats + 1 * 512;
    float* st2 = stats + 2 * 512;
    const float invM = 1.0f / (float)MROWS;

    zero_stats_kernel<<<6, 256, 0, stream>>>(stats);
    fps_kernel<<<BATCH, 1024, 0, stream>>>(xyz, out_xyz);
    knn_kernel<<<BATCH * NPOINT, 32, 0, stream>>>(xyz, out_xyz, idx);
    pad_w0_kernel<<<(C1 * K0PAD + 255) / 256, 256, 0, stream>>>(w0, w0p);
    build_x0_kernel<<<MROWS / 8, 256, 0, stream>>>(xyz, points, out_xyz, idx, x0);

    // layer 1: y1 = x0 * w0^T + b0   (no input BN)
    gemm_wmma_kernel<K0PAD / 4, false><<<dim3(512, C1 / 16), 256, 0, stream>>>(
        x0, K0PAD, w0p, K0PAD, b0, nullptr, nullptr, y1, C1, st0, st0 + 128);
    finalize_stats_kernel<<<1, C1, 0, stream>>>(st0, st0 + 128, g0, be0,
                                                st0 + 256, st0 + 384, C1, invM);

    // layer 2: y2 = relu(bn(y1)) * w1^T + b1   (BN+ReLU fused into A-load)
    gemm_wmma_kernel<C1 / 4, true><<<dim3(512, C2 / 16), 256, 0, stream>>>(
        y1, C1, w1, C1, b1, st0 + 256, st0 + 384, y2, C2, st1, st1 + 128);
    finalize_stats_kernel<<<1, C2, 0, stream>>>(st1, st1 + 128, g1, be1,
                                                st1 + 256, st1 + 384, C2, invM);

    // layer 3: y3 = relu(bn(y2)) * w2^T + b2
    gemm_wmma_kernel<C2 / 4, true><<<dim3(512, C3 / 16), 256, 0, stream>>>(
        y2, C2, w2, C2, b2, st1 + 256, st1 + 384, y3, C3, st2, st2 + 128);
    finalize_stats_kernel<<<1, C3, 0, stream>>>(st2, st2 + 128, g2, be2,
                                                st2 + 256, st2 + 384, C3, invM);

    maxpool_out_kernel<<<BATCH * NPOINT, C3, 0, stream>>>(y3, st2 + 256, st2 + 384, out_pts);
}